// SelAggNet_14602888806822
// MI455X (gfx1250) — compile-verified
//
#include <hip/hip_runtime.h>
#include <hip/hip_bf16.h>

// SelAggNet: y[b, c] = sum_h out[b, sel[b], h] * W[c, h] + bias[c]
// B=16384, N_COLS=24, HIDDEN=768, C=6. All fp32.
//
// Strategy: memory-bound (AI ~ 3 flop/byte). One wave32 per 16-row tile using
// V_WMMA_F32_16X16X4_F32 (exact fp32). W staged in LDS once per block.

#define BATCH   16384
#define NCOLS   24
#define HIDDEN  768
#define COUT    6

typedef float v2f __attribute__((ext_vector_type(2)));
typedef float v8f __attribute__((ext_vector_type(8)));

__global__ __launch_bounds__(256)
void selagg_wmma_f32_kernel(const float* __restrict__ src,   // [B, NCOLS, HIDDEN]
                            const int*   __restrict__ sel,   // [B]
                            const float* __restrict__ W,     // [COUT, HIDDEN]
                            const float* __restrict__ bias,  // [COUT]
                            float*       __restrict__ y)     // [B, COUT]
{
    __shared__ float sW[COUT * HIDDEN];   // 18 KB

    const int t = threadIdx.x;

    // Cooperative stage of W into LDS (4608 floats / 256 threads = 18 each).
    #pragma unroll
    for (int i = t; i < COUT * HIDDEN; i += 256) {
        sW[i] = W[i];
    }
    __syncthreads();

    const int wave = t >> 5;                  // 0..7
    const int lane = t & 31;
    const int tile = blockIdx.x * 8 + wave;   // 0..1023 (16384/16 tiles)

    const int m    = lane & 15;               // A row / D column index
    const int half = lane >> 4;               // 0: K={0,1}, 1: K={2,3} within chunk

    // --- A: gathered row pointer for this lane's matrix row ---
    const int   row  = tile * 16 + m;
    const int   col  = sel[row];
    const float* arow = src + ((size_t)row * NCOLS + (size_t)col) * HIDDEN + half * 2;

    // --- B: W^T column n = lane&15 (clamped; n>=6 columns are discarded) ---
    const int   n    = (m < COUT) ? m : 0;
    const float* brow = &sW[n * HIDDEN + half * 2];

    v8f acc = {};

    // K sweep: 768 = 192 chunks of 4.  Per chunk: one global b64 + one ds b64
    // + one v_wmma_f32_16x16x4_f32.  Unroll for load ILP.
    #pragma unroll 8
    for (int kk = 0; kk < HIDDEN; kk += 4) {
        v2f a = *(const v2f*)(arow + kk);
        v2f b = *(const v2f*)(brow + kk);
        // (neg_a, A, neg_b, B, c_mod, C, reuse_a, reuse_b)
        acc = __builtin_amdgcn_wmma_f32_16x16x4_f32(
            false, a, false, b, (short)0, acc, false, false);
    }

    // D layout: lane L holds column n = L&15; VGPR v holds row v + 8*(L>>4).
    if (m < COUT) {
        const float bn = bias[m];
        #pragma unroll
        for (int v = 0; v < 8; ++v) {
            const int rowOut = tile * 16 + v + 8 * half;
            y[(size_t)rowOut * COUT + m] = acc[v] + bn;
        }
    }
}

extern "C" void kernel_launch(void* const* d_in, const int* in_sizes, int n_in,
                              void* d_out, int out_size, void* d_ws, size_t ws_size,
                              hipStream_t stream)
{
    const float* src  = (const float*)d_in[0];   // out: [B, NCOLS, HIDDEN] f32
    const int*   sel  = (const int*)  d_in[1];   // sel_cols: [B] (int per harness ABI)
    const float* W    = (const float*)d_in[2];   // [COUT, HIDDEN] f32
    const float* bias = (const float*)d_in[3];   // [COUT] f32
    float*       y    = (float*)d_out;           // [B, COUT] f32

    (void)in_sizes; (void)n_in; (void)out_size; (void)d_ws; (void)ws_size;

    dim3 grid(BATCH / 16 / 8);   // 1024 tiles, 8 waves per block -> 128 blocks
    dim3 block(256);
    hipLaunchKernelGGL(selagg_wmma_f32_kernel, grid, block, 0, stream,
                       src, sel, W, bias, y);
}